// Model_81879256531333
// MI455X (gfx1250) — compile-verified
//
#include <hip/hip_runtime.h>
#include <math.h>

// MI455X / gfx1250, wave32.
// Dmat[i,j] = sb * phi( F[i,:] . G[j,:] ),  F,G per-row 5-vectors from X.
// Rank-5 signed Gram -> two V_WMMA_F32_16X16X4_F32 per 16x16 tile.
// Elementwise map uses custom reduced-range acosh/acos (1 sqrt + 1 hw-log per
// element) instead of libm, since the kernel is VALU-bound over the 604 MB
// non-temporal output stream (26 us store floor @ 23.3 TB/s).

typedef __attribute__((ext_vector_type(2))) float v2f;
typedef __attribute__((ext_vector_type(8))) float v8f;

__device__ __forceinline__ void row_feat(const float* __restrict__ X, int r,
                                         float inv_sb, float f[5]) {
    const float* p = X + (size_t)r * 5;
    float x0 = p[0], x1 = p[1], x2 = p[2], x3 = p[3], x4 = p[4];
    float s0 = x3 * inv_sb;
    float s1 = x4 * inv_sb;
    float tn = sqrtf(x0 * x0 + x1 * x1 + x2 * x2);
    tn = fmaxf(tn, 1e-12f);
    float scale = sqrtf(s0 * s0 + s1 * s1 + 1.0f) / tn;  // sqrt(space_norm+1)/|Xt|
    f[0] = s0;
    f[1] = s1;
    f[2] = scale * x0;
    f[3] = scale * x1;
    f[4] = scale * x2;
}

__global__ __launch_bounds__(256) void lorentz_dist_wmma(
    const float* __restrict__ X, const float* __restrict__ sqrtbeta,
    float* __restrict__ out, int N, int tilesPerRow) {
    const int tid  = threadIdx.x;
    const int lane = tid & 31;
    const int wave = tid >> 5;

    const int tileJ = blockIdx.x * 8 + wave;  // wave-uniform
    const int tileI = blockIdx.y;
    if (tileJ >= tilesPerRow) return;         // wave-uniform exit; EXEC stays all-ones

    const int i0 = tileI * 16;
    const int j0 = tileJ * 16;

    // softplus(sqrtbeta) -- same scalar in every lane
    const float sbv    = sqrtbeta[0];
    const float sb     = log1pf(__expf(sbv));
    const float inv_sb = 1.0f / sb;

    // Per-lane: full 5-vector features for row (i0 + m) and col (j0 + m), m = lane&15.
    const int m = lane & 15;
    float F[5], G[5];
    row_feat(X, i0 + m, inv_sb, F);
    row_feat(X, j0 + m, inv_sb, G);
    G[2] = -G[2];
    G[3] = -G[3];
    G[4] = -G[4];

    // A 16x4 f32 fragment: lane = m + 16*kh, v2f = {K=2kh, K=2kh+1}.
    // B 4x16 f32 fragment: lane = n + 16*kh, v2f = {K=2kh, K=2kh+1}.
    const int kh = lane >> 4;
    v2f a0, b0, a1, b1;
    a0.x = kh ? F[2] : F[0];
    a0.y = kh ? F[3] : F[1];
    b0.x = kh ? G[2] : G[0];
    b0.y = kh ? G[3] : G[1];
    // second WMMA covers k=4 (k=5..7 zero padded)
    a1.x = kh ? 0.0f : F[4];
    a1.y = 0.0f;
    b1.x = kh ? 0.0f : G[4];
    b1.y = 0.0f;

    v8f c = {};
    c = __builtin_amdgcn_wmma_f32_16x16x4_f32(false, a0, false, b0, (short)0, c,
                                              false, false);
    c = __builtin_amdgcn_wmma_f32_16x16x4_f32(false, a1, false, b1, (short)0, c,
                                              false, false);

    // C/D layout: lane holds col n = lane&15; VGPR v holds row (8*(lane>>4) + v).
    const float eps   = 1e-5f;
    const float PIO2  = 1.57079632679489662f;
    const int n       = lane & 15;
    const int mbase   = (lane >> 4) * 8;
    size_t base = (size_t)(i0 + mbase) * (size_t)N + (size_t)(j0 + n);

#pragma unroll
    for (int v = 0; v < 8; ++v) {
        float Kv   = c[v];
        float negK = -Kv;
        bool hyp    = Kv < (-1.0f - eps);
        bool lowish = Kv < (-1.0f + eps);
        bool pos    = Kv >= 0.0f;
        bool sph    = (!pos) && (!lowish);

        // --- sph path input: x in [0, 1) ---
        float x   = sph ? negK : 0.0f;
        bool  big = x > 0.5f;
        float z   = big ? __builtin_fmaf(-0.5f, x, 0.5f) : x * x;  // acos poly arg

        // one shared sqrt: hyp -> sqrt(negK^2-1); sph/big -> sqrt((1-x)/2); else sqrt(x^2)
        float sq_arg = hyp ? __builtin_fmaf(negK, negK, -1.0f) : z;
        float s      = sqrtf(sq_arg);

        // hyp: acosh(negK) = log(negK + sqrt(negK^2 - 1)), hw log, no division
        float d_hyp = __logf(hyp ? (negK + s) : 1.0f);

        // sph: acos(x) via minimax asin on [0, 0.5] + half-angle fold
        float p = 4.2163199048e-2f;
        p = __builtin_fmaf(p, z, 2.4181311049e-2f);
        p = __builtin_fmaf(p, z, 4.5470025998e-2f);
        p = __builtin_fmaf(p, z, 7.4953002686e-2f);
        p = __builtin_fmaf(p, z, 1.6666752422e-1f);
        float r      = big ? s : x;
        float asin_r = __builtin_fmaf(r * z, p, r);
        float d_sph  = big ? (2.0f * asin_r) : (PIO2 - asin_r);

        float d_euc = fabsf(2.0f * (1.0f + Kv));
        float d_pos = PIO2 + Kv;

        float d = hyp ? d_hyp : (lowish ? d_euc : (pos ? d_pos : d_sph));
        __builtin_nontemporal_store(sb * d, out + base + (size_t)v * (size_t)N);
    }
}

extern "C" void kernel_launch(void* const* d_in, const int* in_sizes, int n_in,
                              void* d_out, int out_size, void* d_ws, size_t ws_size,
                              hipStream_t stream) {
    (void)n_in; (void)out_size; (void)d_ws; (void)ws_size;
    const float* X        = (const float*)d_in[0];
    const float* sqrtbeta = (const float*)d_in[1];
    float* out            = (float*)d_out;

    const int N     = in_sizes[0] / 5;   // 12288 (multiple of 16)
    const int tiles = N / 16;            // 768

    dim3 block(256);                      // 8 waves -> 8 j-tiles per block
    dim3 grid((tiles + 7) / 8, tiles);    // (96, 768)
    lorentz_dist_wmma<<<grid, block, 0, stream>>>(X, sqrtbeta, out, N, tiles);
}